// BatchedGAT_87368224735381
// MI455X (gfx1250) — compile-verified
//
#include <hip/hip_runtime.h>
#include <hip/hip_bf16.h>

// ---------------- problem constants (match reference) ----------------
constexpr int B_BATCH = 4;
constexpr int N_NODES = 8192;
constexpr int E_EDGES = 131072;
constexpr int ETOT    = E_EDGES + N_NODES;   // self-loops appended
constexpr int IN_DIM  = 128;
constexpr int HEADS   = 4;
constexpr int CH      = 32;
constexpr int OUT_DIM = HEADS * CH;          // 128
constexpr int BN      = B_BATCH * N_NODES;   // 32768

typedef float v2f __attribute__((ext_vector_type(2)));
typedef float v8f __attribute__((ext_vector_type(8)));

// order-preserving float<->uint mapping for atomicMax on floats
__device__ __forceinline__ unsigned fkey(float f) {
    unsigned u = __float_as_uint(f);
    return (u & 0x80000000u) ? ~u : (u | 0x80000000u);
}
__device__ __forceinline__ float fdec(unsigned u) {
    unsigned b = (u & 0x80000000u) ? (u & 0x7fffffffu) : ~u;
    return __uint_as_float(b);
}
__device__ __forceinline__ float lrelu02(float v) {
    return v > 0.0f ? v : 0.2f * v;
}

// ---------------------------------------------------------------------
// Kernel 1: xl = x@W_l + b_l ; xr = x@W_r + b_r   via V_WMMA_F32_16X16X4_F32
// One wave per 16x16 output tile. wave -> (rowTile, colTile, which-matrix).
// ---------------------------------------------------------------------
__global__ __launch_bounds__(256) void gat_gemm_wmma(
    const float* __restrict__ x,
    const float* __restrict__ Wl, const float* __restrict__ bl,
    const float* __restrict__ Wr, const float* __restrict__ br,
    float* __restrict__ xl, float* __restrict__ xr)
{
    const int lane   = threadIdx.x & 31;
    const int wave   = blockIdx.x * 8 + (threadIdx.x >> 5);
    const int mat     = wave & 1;            // 0 -> W_l, 1 -> W_r
    const int colTile = (wave >> 1) & 7;     // 8 tiles of 16 cols
    const int rowTile = wave >> 4;           // BN/16 = 2048 row tiles

    const int mOrN  = lane & 15;             // M for A, N for B/C/D
    const int khalf = (lane >> 4) * 2;       // K offset by lane half
    const int row   = rowTile * 16 + mOrN;
    const int col   = colTile * 16 + mOrN;

    const float* __restrict__ W    = mat ? Wr : Wl;
    const float* __restrict__ bias = mat ? br : bl;
    float*       __restrict__ out  = mat ? xr : xl;

    v8f acc = {};
#if __has_builtin(__builtin_amdgcn_wmma_f32_16x16x4_f32)
    #pragma unroll
    for (int k = 0; k < IN_DIM; k += 4) {
        // A: 16x4 tile of x.  lane holds (M=row, K=k+khalf+{0,1}) -> float2
        v2f a = *(const v2f*)(x + (size_t)row * IN_DIM + k + khalf);
        // B: 4x16 tile of W.  lane holds (K=k+khalf+{0,1}, N=col)
        v2f b;
        b.x = W[(size_t)(k + khalf + 0) * OUT_DIM + col];
        b.y = W[(size_t)(k + khalf + 1) * OUT_DIM + col];
        acc = __builtin_amdgcn_wmma_f32_16x16x4_f32(
                  false, a, false, b, (short)0, acc, false, false);
    }
#else
    // scalar fallback (should not be hit on gfx1250)
    const int rh = (lane >> 4) * 8;
    #pragma unroll
    for (int v = 0; v < 8; ++v) {
        float s = 0.0f;
        const int r = rowTile * 16 + rh + v;
        for (int k = 0; k < IN_DIM; ++k)
            s += x[(size_t)r * IN_DIM + k] * W[(size_t)k * OUT_DIM + col];
        acc[v] = s;
    }
#endif
    // C/D layout: VGPR v holds (M = v + 8*(lane>=16), N = lane&15)
    const int rowHalf = (lane >> 4) * 8;
    const float bv = bias[col];
    #pragma unroll
    for (int v = 0; v < 8; ++v)
        out[(size_t)(rowTile * 16 + rowHalf + v) * OUT_DIM + col] = acc[v] + bv;
}

// ---------------------------------------------------------------------
// Kernel 2: per-edge attention logits + segment max (one wave per edge)
// e[e,h] = sum_c leaky_relu(xl[src,h,c] + xr[dst,h,c]) * att[h,c]
// ---------------------------------------------------------------------
__global__ __launch_bounds__(256) void gat_edge_logits(
    const int* __restrict__ eidx,
    const float* __restrict__ xl, const float* __restrict__ xr,
    const float* __restrict__ att,
    float* __restrict__ e, unsigned* __restrict__ m_u)
{
    const int lane = threadIdx.x & 31;
    const int wave = blockIdx.x * 8 + (threadIdx.x >> 5);
    const int b    = wave / ETOT;
    const int ed   = wave - b * ETOT;
    if (b >= B_BATCH) return;

    int src, dst;
    if (ed < E_EDGES) {
        src = eidx[((size_t)b * 2 + 0) * E_EDGES + ed];
        dst = eidx[((size_t)b * 2 + 1) * E_EDGES + ed];
    } else {
        src = dst = ed - E_EDGES;            // self loop
    }

    const float4 a4 = *(const float4*)(xl + ((size_t)b * N_NODES + src) * OUT_DIM + lane * 4);
    const float4 r4 = *(const float4*)(xr + ((size_t)b * N_NODES + dst) * OUT_DIM + lane * 4);
    const float4 t4 = *(const float4*)(att + lane * 4);

    float p = lrelu02(a4.x + r4.x) * t4.x
            + lrelu02(a4.y + r4.y) * t4.y
            + lrelu02(a4.z + r4.z) * t4.z
            + lrelu02(a4.w + r4.w) * t4.w;
    // reduce within each 8-lane group (one head per group; C=32 = 8 lanes * 4)
    p += __shfl_xor(p, 1, 32);
    p += __shfl_xor(p, 2, 32);
    p += __shfl_xor(p, 4, 32);

    if ((lane & 7) == 0) {
        const int h = lane >> 3;
        e[((size_t)b * ETOT + ed) * HEADS + h] = p;
        atomicMax(&m_u[((size_t)b * N_NODES + dst) * HEADS + h], fkey(p));
    }
}

// ---------------------------------------------------------------------
// Kernel 3: ex = exp(e - m[dst]); denom[dst,h] += ex; accum[dst,:] += ex*xl[src,:]
// (softmax-then-sum == numerator/denominator accumulated separately)
// ---------------------------------------------------------------------
__global__ __launch_bounds__(256) void gat_edge_scatter(
    const int* __restrict__ eidx,
    const float* __restrict__ xl, const float* __restrict__ e,
    const unsigned* __restrict__ m_u,
    float* __restrict__ denom, float* __restrict__ accum)
{
    const int lane = threadIdx.x & 31;
    const int wave = blockIdx.x * 8 + (threadIdx.x >> 5);
    const int b    = wave / ETOT;
    const int ed   = wave - b * ETOT;
    if (b >= B_BATCH) return;

    int src, dst;
    if (ed < E_EDGES) {
        src = eidx[((size_t)b * 2 + 0) * E_EDGES + ed];
        dst = eidx[((size_t)b * 2 + 1) * E_EDGES + ed];
    } else {
        src = dst = ed - E_EDGES;
    }

    const int h    = lane >> 3;
    const float eh = e[((size_t)b * ETOT + ed) * HEADS + h];
    const float mh = fdec(m_u[((size_t)b * N_NODES + dst) * HEADS + h]);
    const float ex = __expf(eh - mh);

    if ((lane & 7) == 0)
        atomicAdd(&denom[((size_t)b * N_NODES + dst) * HEADS + h], ex);

    const float4 a4 = *(const float4*)(xl + ((size_t)b * N_NODES + src) * OUT_DIM + lane * 4);
    float* ab = accum + ((size_t)b * N_NODES + dst) * OUT_DIM + lane * 4;
    atomicAdd(ab + 0, ex * a4.x);
    atomicAdd(ab + 1, ex * a4.y);
    atomicAdd(ab + 2, ex * a4.z);
    atomicAdd(ab + 3, ex * a4.w);
}

// ---------------------------------------------------------------------
// Kernel 4: out = LN(accum/denom + bias) * gamma + beta  (one wave / node)
// ---------------------------------------------------------------------
__global__ __launch_bounds__(256) void gat_finalize(
    const float* __restrict__ accum, const float* __restrict__ denom,
    const float* __restrict__ bias, const float* __restrict__ gamma,
    const float* __restrict__ beta, float* __restrict__ out)
{
    const int lane = threadIdx.x & 31;
    const int node = blockIdx.x * 8 + (threadIdx.x >> 5);   // over BN
    const float4 a4 = *(const float4*)(accum + (size_t)node * OUT_DIM + lane * 4);
    const float  dh = denom[(size_t)node * HEADS + (lane >> 3)] + 1e-16f;
    const float4 bi = *(const float4*)(bias + lane * 4);

    float v0 = a4.x / dh + bi.x;
    float v1 = a4.y / dh + bi.y;
    float v2 = a4.z / dh + bi.z;
    float v3 = a4.w / dh + bi.w;

    float s  = v0 + v1 + v2 + v3;
    float sq = v0 * v0 + v1 * v1 + v2 * v2 + v3 * v3;
    #pragma unroll
    for (int i = 1; i < 32; i <<= 1) {
        s  += __shfl_xor(s,  i, 32);
        sq += __shfl_xor(sq, i, 32);
    }
    const float mean = s * (1.0f / OUT_DIM);
    const float var  = sq * (1.0f / OUT_DIM) - mean * mean;
    const float inv  = rsqrtf(var + 1e-5f);

    const float4 g4 = *(const float4*)(gamma + lane * 4);
    const float4 b4 = *(const float4*)(beta + lane * 4);
    float4 o;
    o.x = (v0 - mean) * inv * g4.x + b4.x;
    o.y = (v1 - mean) * inv * g4.y + b4.y;
    o.z = (v2 - mean) * inv * g4.z + b4.z;
    o.w = (v3 - mean) * inv * g4.w + b4.w;
    *(float4*)(out + (size_t)node * OUT_DIM + lane * 4) = o;
}

// ---------------------------------------------------------------------
extern "C" void kernel_launch(void* const* d_in, const int* in_sizes, int n_in,
                              void* d_out, int out_size, void* d_ws, size_t ws_size,
                              hipStream_t stream) {
    const float* x     = (const float*)d_in[0];
    const int*   eidx  = (const int*)  d_in[1];
    const float* Wl    = (const float*)d_in[2];
    const float* bl    = (const float*)d_in[3];
    const float* Wr    = (const float*)d_in[4];
    const float* br    = (const float*)d_in[5];
    const float* att   = (const float*)d_in[6];
    const float* bias  = (const float*)d_in[7];
    const float* gamma = (const float*)d_in[8];
    const float* beta  = (const float*)d_in[9];
    float* out = (float*)d_out;

    // workspace partition (floats)
    const size_t szXL    = (size_t)BN * OUT_DIM;          // 4,194,304
    const size_t szE     = (size_t)B_BATCH * ETOT * HEADS;
    const size_t szH     = (size_t)BN * HEADS;
    const size_t needed  = (3 * szXL + szE + 2 * szH) * sizeof(float);
    if (ws_size < needed) return;

    float*    xl    = (float*)d_ws;
    float*    xr    = xl + szXL;
    float*    accum = xr + szXL;
    float*    e     = accum + szXL;
    unsigned* m_u   = (unsigned*)(e + szE);
    float*    denom = (float*)(m_u + szH);

    hipMemsetAsync(accum, 0, szXL * sizeof(float), stream);
    hipMemsetAsync(m_u,   0, szH * sizeof(unsigned), stream);
    hipMemsetAsync(denom, 0, szH * sizeof(float), stream);

    // 1) GEMMs: (BN/16) rowTiles * 8 colTiles * 2 matrices waves, 8 waves/block
    const int gemmWaves = (BN / 16) * 8 * 2;              // 32768
    gat_gemm_wmma<<<gemmWaves / 8, 256, 0, stream>>>(x, Wl, bl, Wr, br, xl, xr);

    // 2) edge logits + segment max: one wave per (batch, edge)
    const int edgeWaves = B_BATCH * ETOT;                 // 557,056
    gat_edge_logits<<<edgeWaves / 8, 256, 0, stream>>>(eidx, xl, xr, att, e, m_u);

    // 3) exp / denom / numerator scatter
    gat_edge_scatter<<<edgeWaves / 8, 256, 0, stream>>>(eidx, xl, e, m_u, denom, accum);

    // 4) bias + LayerNorm
    gat_finalize<<<BN / 8, 256, 0, stream>>>(accum, denom, bias, gamma, beta, out);
}